// HGCN_45621142618224
// MI455X (gfx1250) — compile-verified
//
#include <hip/hip_runtime.h>
#include <hip/hip_bf16.h>

#define D 64
#define C_CONST 1.0f
#define SQRT_C  1.0f
#define O0      1.0f      /* origin time coord = 1/sqrt(C) */
#define EPS     1e-7f

typedef float v8f __attribute__((ext_vector_type(8)));
typedef float v2f __attribute__((ext_vector_type(2)));

__device__ __forceinline__ float waveSum(float v) {
#pragma unroll
  for (int off = 16; off > 0; off >>= 1) v += __shfl_xor(v, off, 32);
  return v;
}

// ---------------- zero workspace ----------------
__global__ void zero_kernel(float* __restrict__ p, long n) {
  long i = (long)blockIdx.x * blockDim.x + threadIdx.x;
  if (i < n) p[i] = 0.0f;
}

// ---------------- edge scatter: out[rows[e]] += w[e] * h[cols[e]] ----------------
__global__ void scatter_kernel(const float* __restrict__ hin,
                               const int* __restrict__ rows,
                               const int* __restrict__ cols,
                               const float* __restrict__ ew,
                               float* __restrict__ out, int E) {
  int lane   = threadIdx.x & 31;
  int wave   = (blockIdx.x * blockDim.x + threadIdx.x) >> 5;
  int nwaves = (gridDim.x * blockDim.x) >> 5;
  for (int e = wave; e < E; e += nwaves) {
    int en = e + nwaves;
    if (en < E) {                      // gfx1250 global_prefetch_b8
      __builtin_prefetch(&rows[en], 0, 0);
      __builtin_prefetch(&cols[en], 0, 0);
      __builtin_prefetch(&ew[en], 0, 0);
    }
    int   r = rows[e];
    int   c = cols[e];
    float w = ew[e];
    float2 hv = ((const float2*)hin)[c * 32 + lane];
    atomicAdd(&out[r * D + 2 * lane],     w * hv.x);
    atomicAdd(&out[r * D + 2 * lane + 1], w * hv.y);
  }
}

// ---------------- proj + hyperboloid normalize (in place), wave per node ----------------
__global__ void projnorm_kernel(float* __restrict__ h, int N) {
  int lane = threadIdx.x & 31;
  int node = (blockIdx.x * blockDim.x + threadIdx.x) >> 5;
  if (node >= N) return;
  float2 v = ((float2*)h)[node * 32 + lane];
  // spatial sum of squares (dims 1..63)
  float sq = (lane == 0 ? 0.0f : v.x * v.x) + v.y * v.y;
  float sp = waveSum(sq);
  float x0 = sqrtf(1.0f / C_CONST + sp);
  if (lane == 0) v.x = x0;
  float mink = (sp + x0 * x0) - 2.0f * x0 * x0;
  float sc   = (1.0f / sqrtf(fabsf(mink))) * (1.0f / SQRT_C);
  v.x *= sc; v.y *= sc;
  ((float2*)h)[node * 32 + lane] = v;
}

// ---------------- column sums via WMMA f32 16x16x4 (A = ones) ----------------
// s[d] = sum_n h[n][d].  A(16x4)=1 => every row of D = column sums of B(4x16) tile.
// Main loop is branchless over full 4-row chunks; tail rows handled by wave 0.
__global__ void colsum_wmma_kernel(const float* __restrict__ h,
                                   float* __restrict__ s, int N) {
  int lane   = threadIdx.x & 31;
  int wave   = (blockIdx.x * blockDim.x + threadIdx.x) >> 5;
  int nwaves = (gridDim.x * blockDim.x) >> 5;
  int half   = lane >> 4;       // 0: K rows {0,1}; 1: K rows {2,3}
  int ln     = lane & 15;       // column within 16-wide tile

  v2f a; a[0] = 1.0f; a[1] = 1.0f;      // all-ones A (16x4)
  v8f acc0 = {}; v8f acc1 = {}; v8f acc2 = {}; v8f acc3 = {};

  int fullChunks = N >> 2;              // 4 nodes per WMMA K step
  for (int ch = wave; ch < fullChunks; ch += nwaves) {
    const float* base = h + ((long)ch * 4 + half * 2) * D + ln;
    v2f b0, b1, b2, b3;                 // 8 straight-line coalesced loads
    b0[0] = base[0];   b0[1] = base[D];
    b1[0] = base[16];  b1[1] = base[D + 16];
    b2[0] = base[32];  b2[1] = base[D + 32];
    b3[0] = base[48];  b3[1] = base[D + 48];
    acc0 = __builtin_amdgcn_wmma_f32_16x16x4_f32(false, a, false, b0, (short)0, acc0, false, false);
    acc1 = __builtin_amdgcn_wmma_f32_16x16x4_f32(false, a, false, b1, (short)0, acc1, false, false);
    acc2 = __builtin_amdgcn_wmma_f32_16x16x4_f32(false, a, false, b2, (short)0, acc2, false, false);
    acc3 = __builtin_amdgcn_wmma_f32_16x16x4_f32(false, a, false, b3, (short)0, acc3, false, false);
  }
  // D VGPR0, lanes 0-15 = row M=0 (all rows identical = column sums)
  if (half == 0) {
    atomicAdd(&s[0 * 16 + ln], acc0[0]);
    atomicAdd(&s[1 * 16 + ln], acc1[0]);
    atomicAdd(&s[2 * 16 + ln], acc2[0]);
    atomicAdd(&s[3 * 16 + ln], acc3[0]);
  }
  // tail (N % 4 rows, at most 3; empty for N=50000)
  if (wave == 0) {
    for (int r = fullChunks * 4; r < N; ++r) {
      atomicAdd(&s[2 * lane],     h[(long)r * D + 2 * lane]);
      atomicAdd(&s[2 * lane + 1], h[(long)r * D + 2 * lane + 1]);
    }
  }
}

// stats layout (floats): [0,64) colsum s; [64] varsum; [72,136) mu; [136] den1; [137] scale
__global__ void centroid_kernel(float* __restrict__ stats, int N) {
  __shared__ float sm[D];
  __shared__ float red[2];
  int t = threadIdx.x;                       // 0..63
  float v = stats[t] / (float)N;             // mean
  sm[t] = v;
  __syncthreads();
  float ws = waveSum(v * v);
  if ((t & 31) == 0) red[t >> 5] = ws;
  __syncthreads();
  float total = red[0] + red[1];
  float s0    = sm[0];
  float mink  = total - 2.0f * s0 * s0;
  float denom = SQRT_C * sqrtf(fabsf(mink) + EPS);
  stats[72 + t] = v / denom;                 // mu
  if (t == 0) {
    float mu0     = s0 / denom;
    float minkmuo = mu0 * O0 - 2.0f * mu0 * O0;
    stats[136]    = 1.0f - C_CONST * minkmuo;   // den of transp(mu -> o)
  }
}

// logmap(mu,x) + transp(mu,o,.) then accumulate sum of ||u||_2
__global__ void logvar_kernel(const float* __restrict__ h,
                              const float* __restrict__ stats,
                              float* __restrict__ varsum, int N) {
  __shared__ float bsum;
  if (threadIdx.x == 0) bsum = 0.0f;
  __syncthreads();
  int lane = threadIdx.x & 31;
  int node = (blockIdx.x * blockDim.x + threadIdx.x) >> 5;
  if (node < N) {
    const float* mu = stats + 72;
    float den1 = stats[136];
    float2 x = ((const float2*)h)[node * 32 + lane];
    float2 m = ((const float2*)mu)[lane];
    float dot = waveSum(x.x * m.x + x.y * m.y);
    float x0  = __shfl(x.x, 0, 32);
    float mu0 = __shfl(m.x, 0, 32);
    float minkmux = dot - 2.0f * mu0 * x0;
    float alpha = fmaxf(-C_CONST * minkmux, 1.0f + EPS);
    float coef  = acoshf(alpha) / sqrtf(alpha * alpha - 1.0f);
    float2 u;
    u.x = coef * (x.x - alpha * m.x);
    u.y = coef * (x.y - alpha * m.y);
    float u0     = __shfl(u.x, 0, 32);
    float minkou = O0 * u0 - 2.0f * O0 * u0;
    float f      = (C_CONST * minkou) / den1;
    u.x += f * m.x + (lane == 0 ? f * O0 : 0.0f);
    u.y += f * m.y;
    float ss = waveSum(u.x * u.x + u.y * u.y);
    if (lane == 0) atomicAdd(&bsum, sqrtf(ss));
  }
  __syncthreads();
  if (threadIdx.x == 0) atomicAdd(varsum, bsum);
}

__global__ void finalize_kernel(float* __restrict__ stats,
                                const float* __restrict__ gamma, int N) {
  float var  = stats[64] / (float)N;
  stats[137] = gamma[0] / (var + EPS);
}

// recompute u, scale by gamma/(var+eps), transp(o,o,.), expmap(o,.) -> out
__global__ void expmap_kernel(const float* __restrict__ h,
                              const float* __restrict__ stats,
                              float* __restrict__ out, int N) {
  int lane = threadIdx.x & 31;
  int node = (blockIdx.x * blockDim.x + threadIdx.x) >> 5;
  if (node >= N) return;
  const float* mu = stats + 72;
  float den1  = stats[136];
  float scale = stats[137];
  float2 x = ((const float2*)h)[node * 32 + lane];
  float2 m = ((const float2*)mu)[lane];
  float dot = waveSum(x.x * m.x + x.y * m.y);
  float x0  = __shfl(x.x, 0, 32);
  float mu0 = __shfl(m.x, 0, 32);
  float minkmux = dot - 2.0f * mu0 * x0;
  float alpha = fmaxf(-C_CONST * minkmux, 1.0f + EPS);
  float coef  = acoshf(alpha) / sqrtf(alpha * alpha - 1.0f);
  float2 u;
  u.x = coef * (x.x - alpha * m.x);
  u.y = coef * (x.y - alpha * m.y);
  float u0     = __shfl(u.x, 0, 32);
  float minkou = O0 * u0 - 2.0f * O0 * u0;
  float f      = (C_CONST * minkou) / den1;
  u.x += f * m.x + (lane == 0 ? f * O0 : 0.0f);
  u.y += f * m.y;
  // u *= gamma/(var+eps)
  u.x *= scale; u.y *= scale;
  // transp(o,o,u): den = 1 - C*mink(o,o) = 2; only dim0 affected
  float u0s     = __shfl(u.x, 0, 32);
  float minkou2 = O0 * u0s - 2.0f * O0 * u0s;
  float minkoo  = O0 * O0 - 2.0f * O0 * O0;
  float f2      = (C_CONST * minkou2) / (1.0f - C_CONST * minkoo);
  if (lane == 0) u.x += f2 * (2.0f * O0);
  // expmap(o,u)
  float u0n    = __shfl(u.x, 0, 32);
  float ssq    = waveSum(u.x * u.x + u.y * u.y);
  float minkuu = ssq - 2.0f * u0n * u0n;
  float nrm    = sqrtf(fmaxf(minkuu, EPS));
  float theta  = SQRT_C * nrm;
  float sh     = sinhf(theta) / theta;
  float2 r;
  r.x = sh * u.x + (lane == 0 ? coshf(theta) * O0 : 0.0f);
  r.y = sh * u.y;
  ((float2*)out)[node * 32 + lane] = r;
}

extern "C" void kernel_launch(void* const* d_in, const int* in_sizes, int n_in,
                              void* d_out, int out_size, void* d_ws, size_t ws_size,
                              hipStream_t stream) {
  const float* x     = (const float*)d_in[0];
  const int*   rows  = (const int*)d_in[1];
  const int*   cols  = (const int*)d_in[2];
  const float* ew    = (const float*)d_in[3];
  const float* gamma = (const float*)d_in[4];
  int N = in_sizes[0] / D;
  int E = in_sizes[1];

  float* hout   = (float*)d_out;
  float* wsA    = (float*)d_ws;               // N*D accumulator / working h
  float* stats  = wsA + (size_t)N * D;        // 160 floats of stats
  float* varsum = stats + 64;

  long zeroN      = (long)N * D + 160;
  int  zeroBlocks = (int)((zeroN + 255) / 256);
  int  nodeBlocks = (N * 32 + 255) / 256;     // wave per node, 8 nodes/block

  for (int layer = 0; layer < 2; ++layer) {
    const float* hin = (layer == 0) ? x : hout;
    zero_kernel   <<<zeroBlocks, 256, 0, stream>>>(wsA, zeroN);
    scatter_kernel<<<4096, 256, 0, stream>>>(hin, rows, cols, ew, wsA, E);
    projnorm_kernel<<<nodeBlocks, 256, 0, stream>>>(wsA, N);
    colsum_wmma_kernel<<<64, 256, 0, stream>>>(wsA, stats, N);
    centroid_kernel<<<1, 64, 0, stream>>>(stats, N);
    logvar_kernel <<<nodeBlocks, 256, 0, stream>>>(wsA, stats, varsum, N);
    finalize_kernel<<<1, 1, 0, stream>>>(stats, gamma, N);
    expmap_kernel <<<nodeBlocks, 256, 0, stream>>>(wsA, stats, hout, N);
  }
}